// GCN2Net_69002944578216
// MI455X (gfx1250) — compile-verified
//
#include <hip/hip_runtime.h>
#include <math.h>

#define N_NODES 100000
#define E_EDGES 1600000
#define F_IN  256
#define F_H   128
#define F_OUT 64
#define N_LAYERS 4
#define ALPHA 0.5f

typedef __attribute__((ext_vector_type(2))) float v2f;
typedef __attribute__((ext_vector_type(8))) float v8f;

// ---------------- CSR construction ----------------

__global__ void init_deg_kernel(int* deg, int n) {
    int i = blockIdx.x * blockDim.x + threadIdx.x;
    if (i < n) deg[i] = 1;                       // self-loop
}

__global__ void count_deg_kernel(const int* __restrict__ row, int* deg, int e) {
    int i = blockIdx.x * blockDim.x + threadIdx.x;
    if (i < e) atomicAdd(&deg[row[i]], 1);
}

// single-block exclusive scan (N=100000, trivial runtime) + dinv = rsqrt(deg)
__global__ void scan_kernel(const int* __restrict__ deg, int* row_ptr, float* dinv, int n) {
    __shared__ int sh[1024];
    int tid = threadIdx.x;
    int running = 0;
    for (int base = 0; base < n; base += 1024) {
        int i = base + tid;
        int v = (i < n) ? deg[i] : 0;
        if (i < n) dinv[i] = rsqrtf((float)v);   // deg >= 1 always
        sh[tid] = v;
        __syncthreads();
        for (int off = 1; off < 1024; off <<= 1) {
            int t = (tid >= off) ? sh[tid - off] : 0;
            __syncthreads();
            sh[tid] += t;
            __syncthreads();
        }
        if (i < n) row_ptr[i] = running + sh[tid] - v;   // exclusive
        running += sh[1023];
        __syncthreads();
    }
    if (tid == 0) row_ptr[n] = running;
}

__global__ void copy_fill_kernel(const int* __restrict__ row_ptr, int* fill, int n) {
    int i = blockIdx.x * blockDim.x + threadIdx.x;
    if (i < n) fill[i] = row_ptr[i];
}

__global__ void scatter_edges_kernel(const int* __restrict__ row, const int* __restrict__ col,
                                     const float* __restrict__ dinv, int* fill,
                                     int* csr_col, float* csr_nrm, int e) {
    int i = blockIdx.x * blockDim.x + threadIdx.x;
    if (i < e) {
        int r = row[i], c = col[i];
        int pos = atomicAdd(&fill[r], 1);
        csr_col[pos] = c;
        csr_nrm[pos] = dinv[r] * dinv[c];
    }
}

__global__ void scatter_loops_kernel(const float* __restrict__ dinv, int* fill,
                                     int* csr_col, float* csr_nrm, int n) {
    int i = blockIdx.x * blockDim.x + threadIdx.x;
    if (i < n) {
        int pos = atomicAdd(&fill[i], 1);
        csr_col[pos] = i;
        csr_nrm[pos] = dinv[i] * dinv[i];
    }
}

// ---------------- sparse propagation (L2-resident gather) ----------------
// one wave32 per node; lane owns 4 contiguous features (float4); fused
// initial residual: hid = (1-alpha)*agg + alpha*x0
__global__ void spmm_kernel(const int* __restrict__ row_ptr, const int* __restrict__ csr_col,
                            const float* __restrict__ csr_nrm, const float* __restrict__ hin,
                            const float* __restrict__ x0, float* __restrict__ hid, int n) {
    int wid  = (blockIdx.x * blockDim.x + threadIdx.x) >> 5;
    int lane = threadIdx.x & 31;
    if (wid >= n) return;
    int s = row_ptr[wid], e = row_ptr[wid + 1];
    const int fo = lane * 4;
    float4 acc; acc.x = 0.f; acc.y = 0.f; acc.z = 0.f; acc.w = 0.f;
    for (int i = s; i < e; ++i) {
        int   c = csr_col[i];
        float w = csr_nrm[i];
        if (i + 1 < e)
            __builtin_prefetch(hin + (size_t)csr_col[i + 1] * F_H + fo, 0, 0);
        float4 hv = *(const float4*)(hin + (size_t)c * F_H + fo);
        acc.x += w * hv.x; acc.y += w * hv.y; acc.z += w * hv.z; acc.w += w * hv.w;
    }
    float4 xv = *(const float4*)(x0 + (size_t)wid * F_H + fo);
    float4 o;
    o.x = (1.0f - ALPHA) * acc.x + ALPHA * xv.x;
    o.y = (1.0f - ALPHA) * acc.y + ALPHA * xv.y;
    o.z = (1.0f - ALPHA) * acc.z + ALPHA * xv.z;
    o.w = (1.0f - ALPHA) * acc.w + ALPHA * xv.w;
    *(float4*)(hid + (size_t)wid * F_H + fo) = o;
}

// ---------------- WMMA f32 GEMM: C[M x NN] = A[M x K] @ B[K x NN] (+ epilogue) ----
// mode 0: C = A@B + bias        (bias broadcast over rows)
// mode 1: C = relu((1-beta)*hid + beta*(A@B))     (GCNII identity mapping)
// One wave per 16-row M tile, covering all NTILES 16-col N tiles.
// B pre-packed in LDS as float2 pairs (B[k][n], B[k+1][n]) to match the
// V_WMMA_F32_16X16X4_F32 per-lane layout: lanes 0-15 -> K=0,1 ; lanes 16-31 -> K=2,3.
template <int NTILES>
__global__ void wmma_gemm_kernel(const float* __restrict__ A, const float* __restrict__ B,
                                 const float* __restrict__ bias, const float* __restrict__ hid,
                                 float* __restrict__ C, int M, int K, int mode, float beta) {
    constexpr int NN = NTILES * 16;
    extern __shared__ __align__(16) char smem_raw[];
    float2* Bp = (float2*)smem_raw;

    // cooperative pair-pack of B into LDS
    const int tot = (K >> 1) * NN;
    for (int i = threadIdx.x; i < tot; i += blockDim.x) {
        int k2 = i / NN;
        int nn = i - k2 * NN;
        float2 pr;
        pr.x = B[(size_t)(2 * k2) * NN + nn];
        pr.y = B[(size_t)(2 * k2 + 1) * NN + nn];
        Bp[i] = pr;
    }
    __syncthreads();

    int wid  = (blockIdx.x * blockDim.x + threadIdx.x) >> 5;
    int lane = threadIdx.x & 31;
    int mtiles = M >> 4;
    if (wid >= mtiles) return;   // whole-wave uniform; EXEC all-1s for active waves

    const int row0  = wid << 4;
    const int ln15  = lane & 15;   // A: row-in-tile ; B/C: col-in-tile
    const int khalf = lane >> 4;   // K pair select (0 -> K=0,1 ; 1 -> K=2,3)

    v8f acc[NTILES];
#pragma unroll
    for (int j = 0; j < NTILES; ++j) {
        float bv = (mode == 0) ? bias[j * 16 + ln15] : 0.0f;
#pragma unroll
        for (int v = 0; v < 8; ++v) acc[j][v] = bv;
    }

    for (int kk = 0; kk < K; kk += 4) {
        v2f av = *(const v2f*)(A + (size_t)(row0 + ln15) * K + kk + 2 * khalf);
        const int kp = (kk >> 1) + khalf;
#pragma unroll
        for (int j = 0; j < NTILES; ++j) {
            float2 bp = Bp[kp * NN + j * 16 + ln15];
            v2f bv; bv.x = bp.x; bv.y = bp.y;
            acc[j] = __builtin_amdgcn_wmma_f32_16x16x4_f32(
                false, av, false, bv, (short)0, acc[j], false, false);
        }
    }

#pragma unroll
    for (int j = 0; j < NTILES; ++j) {
#pragma unroll
        for (int v = 0; v < 8; ++v) {
            int r    = row0 + v + (khalf << 3);   // C/D layout: vgpr v -> M=v / v+8
            int cidx = j * 16 + ln15;
            float val = acc[j][v];
            if (mode == 1) {
                float hv = hid[(size_t)r * NN + cidx];
                val = (1.0f - beta) * hv + beta * val;
                val = val > 0.0f ? val : 0.0f;
            }
            C[(size_t)r * NN + cidx] = val;
        }
    }
}

// ---------------- host launch ----------------

extern "C" void kernel_launch(void* const* d_in, const int* in_sizes, int n_in,
                              void* d_out, int out_size, void* d_ws, size_t ws_size,
                              hipStream_t stream) {
    (void)in_sizes; (void)n_in; (void)out_size; (void)ws_size;

    const float* x  = (const float*)d_in[0];
    const int*   ei = (const int*)d_in[1];     // [2, E] int
    const float* W1 = (const float*)d_in[2];
    const float* b1 = (const float*)d_in[3];
    const float* Wc = (const float*)d_in[4];   // [L, H, H]
    const float* W2 = (const float*)d_in[5];
    const float* b2 = (const float*)d_in[6];
    float* out = (float*)d_out;

    const int n = N_NODES, e = E_EDGES;
    const int* row = ei;
    const int* col = ei + e;

    char* p = (char*)d_ws;
    auto alloc = [&](size_t bytes) -> char* {
        char* r = p;
        p += (bytes + 255) & ~(size_t)255;
        return r;
    };
    int*   deg     = (int*)  alloc(sizeof(int)   * (size_t)n);
    int*   row_ptr = (int*)  alloc(sizeof(int)   * (size_t)(n + 1));
    int*   fill    = (int*)  alloc(sizeof(int)   * (size_t)n);
    float* dinv    = (float*)alloc(sizeof(float) * (size_t)n);
    int*   csr_col = (int*)  alloc(sizeof(int)   * (size_t)(e + n));
    float* csr_nrm = (float*)alloc(sizeof(float) * (size_t)(e + n));
    float* x0      = (float*)alloc(sizeof(float) * (size_t)n * F_H);
    float* hbuf    = (float*)alloc(sizeof(float) * (size_t)n * F_H);
    float* agg     = (float*)alloc(sizeof(float) * (size_t)n * F_H);

    dim3 b256(256);
    init_deg_kernel<<<(n + 255) / 256, b256, 0, stream>>>(deg, n);
    count_deg_kernel<<<(e + 255) / 256, b256, 0, stream>>>(row, deg, e);
    scan_kernel<<<1, 1024, 0, stream>>>(deg, row_ptr, dinv, n);
    copy_fill_kernel<<<(n + 255) / 256, b256, 0, stream>>>(row_ptr, fill, n);
    scatter_edges_kernel<<<(e + 255) / 256, b256, 0, stream>>>(row, col, dinv, fill,
                                                               csr_col, csr_nrm, e);
    scatter_loops_kernel<<<(n + 255) / 256, b256, 0, stream>>>(dinv, fill, csr_col, csr_nrm, n);

    const int mtiles  = n / 16;              // 6250 (exact)
    const int gblocks = (mtiles + 7) / 8;    // 8 waves per 256-thread block

    // x0 = x @ W1 + b1
    wmma_gemm_kernel<8><<<gblocks, b256, (size_t)F_IN * F_H * sizeof(float), stream>>>(
        x, W1, b1, nullptr, x0, n, F_IN, 0, 0.0f);

    const float* hin = x0;
    for (int l = 0; l < N_LAYERS; ++l) {
        spmm_kernel<<<((size_t)n * 32 + 255) / 256, b256, 0, stream>>>(
            row_ptr, csr_col, csr_nrm, hin, x0, agg, n);
        float beta = logf(1.0f / (float)(l + 1) + 1.0f);   // LAM = 1.0
        wmma_gemm_kernel<8><<<gblocks, b256, (size_t)F_H * F_H * sizeof(float), stream>>>(
            agg, Wc + (size_t)l * F_H * F_H, nullptr, agg, hbuf, n, F_H, 1, beta);
        hin = hbuf;
    }

    // out = h @ W2 + b2
    wmma_gemm_kernel<4><<<gblocks, b256, (size_t)F_H * F_OUT * sizeof(float), stream>>>(
        hbuf, W2, b2, nullptr, out, n, F_H, 0, 0.0f);
}